// HetSynRSNNCell_59399397704147
// MI455X (gfx1250) — compile-verified
//
#include <hip/hip_runtime.h>

// ---------------------------------------------------------------------------
// HetSynRSNNCell forward, MI455X / gfx1250 (wave32).
//
// Memory-bound streaming kernel: ~530 MB of HBM traffic per call
// (read 2x128MB hIsyn, write 2x128MB Isyn; rho/w stay L2-resident).
// One 256-thread block per (b, o) row pair; float4 fully-coalesced rows.
// The 32-lane partial-sum reduction is done with V_WMMA_F32_16X16X4_F32
// (B = all-ones), which gives D[m,n] = partial[m] + partial[m+16] exactly
// in fp32; 7 adds + one xor-16 shuffle finish the wave sum.
// ---------------------------------------------------------------------------

typedef float v2f __attribute__((ext_vector_type(2)));
typedef float v8f __attribute__((ext_vector_type(8)));

#define BB      32
#define N_IN    1024
#define N_OUT   1024
#define THR     1.0f
// exp(-0.001 / 0.02)
#define DECAY_R 0.95122942450071400910f

__device__ __forceinline__ float wave_reduce32_wmma(float partial) {
    // A (16x4 f32, 2 VGPRs): VGPR0 = partial, VGPR1 = 0.
    //   lanes 0-15 : A[m,0]=partial[m],    A[m,1]=0
    //   lanes 16-31: A[m,2]=partial[m+16], A[m,3]=0
    // B (4x16 f32) = all ones  ->  D[m,n] = partial[m] + partial[m+16].
    v2f a; a[0] = partial; a[1] = 0.0f;
    v2f ones; ones[0] = 1.0f; ones[1] = 1.0f;
    v8f c = {};
    c = __builtin_amdgcn_wmma_f32_16x16x4_f32(
        /*neg_a=*/false, a, /*neg_b=*/false, ones,
        /*c_mod=*/(short)0, c, /*reuse_a=*/false, /*reuse_b=*/false);
    // C/D layout: lane n (0-15) VGPR j -> D[j, n];  lane 16+n VGPR j -> D[8+j, n].
    float s = ((c[0] + c[1]) + (c[2] + c[3])) + ((c[4] + c[5]) + (c[6] + c[7]));
    // lanes 0-15 hold sum(partial[0..7]) + sum(partial[16..23]);
    // lanes 16-31 hold sum(partial[8..15]) + sum(partial[24..31]).
    s += __shfl_xor(s, 16, 32);
    return s;
}

__global__ __launch_bounds__(256) void hetsyn_cell_kernel(
    const float* __restrict__ x,          // (B, N_IN)
    const float* __restrict__ hz,         // (B, N_OUT)
    const float* __restrict__ hIsyn_in,   // (B, N_OUT, N_IN)
    const float* __restrict__ hIsyn_rec,  // (B, N_OUT, N_OUT)
    const float* __restrict__ hIr,        // (B, N_OUT)
    const float* __restrict__ w_in,       // (N_OUT, N_IN)
    const float* __restrict__ rho_in,     // (N_OUT, N_IN)
    const float* __restrict__ w_rec,      // (N_OUT, N_OUT)
    const float* __restrict__ rho_rec,    // (N_OUT, N_OUT)
    float* __restrict__ out_z,            // (B, N_OUT)
    float* __restrict__ out_isyn_in,      // (B, N_OUT, N_IN)
    float* __restrict__ out_isyn_rec,     // (B, N_OUT, N_OUT)
    float* __restrict__ out_ir)           // (B, N_OUT)
{
    const int bid = blockIdx.x;           // 0 .. B*N_OUT-1
    const int b   = bid >> 10;            // N_OUT == 1024
    const int o   = bid & (N_OUT - 1);
    const int t   = threadIdx.x;          // 0..255, each covers 4 columns
    const int i4  = t << 2;

    const size_t rowBO = (size_t)bid * N_IN;   // (b*N_OUT + o) * N_IN
    const size_t rowO  = (size_t)o   * N_IN;
    const size_t rowBx = (size_t)b   * N_IN;

    // ---- input path: Isyn_in = rho_in * hIsyn_in + w_in * x[b,:] ----
    const float4 h_in = *(const float4*)(hIsyn_in + rowBO + i4);
    const float4 r_in = *(const float4*)(rho_in   + rowO  + i4);
    const float4 wi   = *(const float4*)(w_in     + rowO  + i4);
    const float4 xv   = *(const float4*)(x        + rowBx + i4);
    float4 s_in;
    s_in.x = fmaf(r_in.x, h_in.x, wi.x * xv.x);
    s_in.y = fmaf(r_in.y, h_in.y, wi.y * xv.y);
    s_in.z = fmaf(r_in.z, h_in.z, wi.z * xv.z);
    s_in.w = fmaf(r_in.w, h_in.w, wi.w * xv.w);
    *(float4*)(out_isyn_in + rowBO + i4) = s_in;
    float partial = (s_in.x + s_in.y) + (s_in.z + s_in.w);

    // ---- recurrent path: Isyn_rec = rho_rec * hIsyn_rec + w_rec * hz[b,:] ----
    const float4 h_rc = *(const float4*)(hIsyn_rec + rowBO + i4);
    const float4 r_rc = *(const float4*)(rho_rec   + rowO  + i4);
    const float4 wr   = *(const float4*)(w_rec     + rowO  + i4);
    const float4 zv   = *(const float4*)(hz        + rowBx + i4);
    float4 s_rc;
    s_rc.x = fmaf(r_rc.x, h_rc.x, wr.x * zv.x);
    s_rc.y = fmaf(r_rc.y, h_rc.y, wr.y * zv.y);
    s_rc.z = fmaf(r_rc.z, h_rc.z, wr.z * zv.z);
    s_rc.w = fmaf(r_rc.w, h_rc.w, wr.w * zv.w);
    *(float4*)(out_isyn_rec + rowBO + i4) = s_rc;
    partial += (s_rc.x + s_rc.y) + (s_rc.z + s_rc.w);

    // ---- wave-level reduction via WMMA (EXEC all ones here) ----
    const float wsum = wave_reduce32_wmma(partial);

    // ---- block-level reduction across 8 waves ----
    __shared__ float lds[8];
    const int wave = t >> 5;
    if ((t & 31) == 0) lds[wave] = wsum;
    __syncthreads();

    if (t == 0) {
        float v = ((lds[0] + lds[1]) + (lds[2] + lds[3])) +
                  ((lds[4] + lds[5]) + (lds[6] + lds[7]));
        const size_t bo = (size_t)b * N_OUT + o;
        const float hzv = hz[bo];
        const float ir  = DECAY_R * hIr[bo] + THR * hzv;
        v -= ir;
        out_ir[bo] = ir;
        out_z[bo]  = ((v - THR) > 0.0f) ? 1.0f : 0.0f;
    }
}

extern "C" void kernel_launch(void* const* d_in, const int* in_sizes, int n_in,
                              void* d_out, int out_size, void* d_ws, size_t ws_size,
                              hipStream_t stream) {
    // setup_inputs() order:
    const float* x         = (const float*)d_in[0];
    const float* hz        = (const float*)d_in[1];
    const float* hIsyn_in  = (const float*)d_in[2];
    const float* hIsyn_rec = (const float*)d_in[3];
    const float* hIr       = (const float*)d_in[4];
    const float* w_in      = (const float*)d_in[5];
    const float* rho_in    = (const float*)d_in[6];
    const float* w_rec     = (const float*)d_in[7];
    const float* rho_rec   = (const float*)d_in[8];

    // Output tuple (z, Isyn_in, Isyn_rec, Ir), concatenated flat, fp32.
    float* out          = (float*)d_out;
    float* out_z        = out;
    float* out_isyn_in  = out_z        + (size_t)BB * N_OUT;
    float* out_isyn_rec = out_isyn_in  + (size_t)BB * N_OUT * N_IN;
    float* out_ir       = out_isyn_rec + (size_t)BB * N_OUT * N_OUT;

    dim3 grid(BB * N_OUT);
    dim3 block(256);
    hetsyn_cell_kernel<<<grid, block, 0, stream>>>(
        x, hz, hIsyn_in, hIsyn_rec, hIr,
        w_in, rho_in, w_rec, rho_rec,
        out_z, out_isyn_in, out_isyn_rec, out_ir);
}